// ProbAttention_18270790877567
// MI455X (gfx1250) — compile-verified
//
#include <hip/hip_runtime.h>
#include <hip/hip_bf16.h>
#include <math.h>

#define B_DIM 8
#define H_DIM 8
#define LQ_DIM 2048
#define LK_DIM 2048
#define D_DIM 64
#define U_TOP 40
#define NTILE 3      // ceil(40/16)
#define NCHUNK 16    // cumsum chunks
#define CHUNK 128    // LK / NCHUNK

typedef __attribute__((ext_vector_type(2))) float v2f;
typedef __attribute__((ext_vector_type(8))) float v8f;

// Wave-relative LDS byte offset from a flat pointer to a __shared__ object.
// (ISA: flat->LDS aperture mapping uses addr[31:0] as the wave-relative LDS
// address, which is exactly what GLOBAL_LOAD_ASYNC_TO_LDS's VDST expects.)
__device__ __forceinline__ unsigned lds_off(const void* p) {
    return (unsigned)(reinterpret_cast<uintptr_t>(p));
}

// XOR-butterfly lane exchange via ds_swizzle_b32 (group-of-32 mode,
// and_mask=0x1f, or_mask=0, xor_mask=m). Masks 1/2/4/8 stay inside each
// 16-lane half, giving a width-16 reduction with immediate-offset swizzles
// instead of ds_bpermute + per-call index math.
#define SWZ_XOR(v, m) \
    __int_as_float(__builtin_amdgcn_ds_swizzle(__float_as_int(v), 0x7C00 | (m)))

// ---------------------------------------------------------------------------
// Kernel 1: M[b,h,q] = max_s dot(Q[b,h,q], K[b,h,idx[q,s]]) - mean_s(dot)
// Layouts: queries/keys are (B, L, H, D).
// ---------------------------------------------------------------------------
__global__ __launch_bounds__(256)
void prob_sample_score_kernel(const float* __restrict__ Q,
                              const float* __restrict__ K,
                              const int* __restrict__ idx,
                              float* __restrict__ M,
                              int u_part) {
    int id = blockIdx.x * 256 + threadIdx.x;           // id = (b*H + h)*LQ + q
    if (id >= B_DIM * H_DIM * LQ_DIM) return;
    int q = id % LQ_DIM;
    int h = (id / LQ_DIM) % H_DIM;
    int b = id / (LQ_DIM * H_DIM);

    const float4* qrow =
        (const float4*)&Q[(((size_t)b * LQ_DIM + q) * H_DIM + h) * D_DIM];
    float4 qv[16];
#pragma unroll
    for (int i = 0; i < 16; ++i) qv[i] = qrow[i];

    float mx = -INFINITY, sum = 0.f;
    for (int s = 0; s < u_part; ++s) {
        int kp = idx[q * u_part + s];
        const float4* krow =
            (const float4*)&K[(((size_t)b * LK_DIM + kp) * H_DIM + h) * D_DIM];
        float d = 0.f;
#pragma unroll
        for (int i = 0; i < 16; ++i) {
            float4 kv = krow[i];
            d += qv[i].x * kv.x + qv[i].y * kv.y + qv[i].z * kv.z + qv[i].w * kv.w;
        }
        mx = fmaxf(mx, d);
        sum += d;
    }
    M[id] = mx - sum / (float)u_part;
}

// ---------------------------------------------------------------------------
// Kernel 2: per (b,h) select the 40 largest M values (indices only; order
// inside the set does not affect the final output).
// ---------------------------------------------------------------------------
__global__ __launch_bounds__(256)
void topk_kernel(const float* __restrict__ M, int* __restrict__ Mtop) {
    __shared__ float vals[LQ_DIM];
    __shared__ float rv[256];
    __shared__ int   ri[256];
    int bh = blockIdx.x;
    int tid = threadIdx.x;

    for (int i = tid; i < LQ_DIM; i += 256) vals[i] = M[(size_t)bh * LQ_DIM + i];
    __syncthreads();

    for (int it = 0; it < U_TOP; ++it) {
        float bv = -INFINITY; int bi = LQ_DIM;
        for (int i = tid; i < LQ_DIM; i += 256) {
            float v = vals[i];
            if (v > bv || (v == bv && i < bi)) { bv = v; bi = i; }
        }
        rv[tid] = bv; ri[tid] = bi;
        __syncthreads();
        for (int s = 128; s > 0; s >>= 1) {
            if (tid < s) {
                float v2 = rv[tid + s]; int i2 = ri[tid + s];
                if (v2 > rv[tid] || (v2 == rv[tid] && i2 < ri[tid])) {
                    rv[tid] = v2; ri[tid] = i2;
                }
            }
            __syncthreads();
        }
        int win = ri[0];
        if (tid == 0) {
            Mtop[bh * U_TOP + it] = win;
            vals[win] = -INFINITY;
        }
        __syncthreads();
    }
}

// ---------------------------------------------------------------------------
// Kernel 3: causal attention for the selected queries, one wave (32 lanes)
// per 16-query tile.  Q*K^T and P*V both use V_WMMA_F32_16X16X4_F32.
// K/V tiles are staged into LDS with GLOBAL_LOAD_ASYNC_TO_LDS_B128 (ASYNCcnt),
// avoiding the VMEM->VGPR->DS round trip.
//   A layout (16x4 f32):  lanes 0-15: M=lane, K={d0,d0+1}; lanes 16-31: K={d0+2,d0+3}
//   B layout (4x16 f32):  VGPR0 rows K={0,2}, VGPR1 rows K={1,3}; N = lane%16
//   C/D layout (16x16):   VGPR r: M = r + 8*(lane/16), N = lane%16
// ---------------------------------------------------------------------------
__global__ __launch_bounds__(32)
void sparse_attn_kernel(const float* __restrict__ Q,
                        const float* __restrict__ K,
                        const float* __restrict__ V,
                        const int* __restrict__ Mtop,
                        float* __restrict__ attn_out) {
    const int bid  = blockIdx.x;
    const int tile = bid % NTILE;
    const int bh   = bid / NTILE;
    const int h    = bh % H_DIM;
    const int b    = bh / H_DIM;
    const int lane = threadIdx.x & 31;
    const int g    = lane >> 4;      // lane half
    const int lr   = lane & 15;

    __shared__ float Qs[16][D_DIM];
    __shared__ float Ks[16][D_DIM];
    __shared__ float Vs[16][D_DIM];
    __shared__ float Ps[16][16];
    __shared__ int   qps[16];

    if (lane < 16) {
        int u = tile * 16 + lane;
        qps[lane] = (u < U_TOP) ? Mtop[bh * U_TOP + u] : -1;
    }
    __syncthreads();

    // Load Q tile (zeros for pad rows) - predicated, so keep the VGPR path.
#pragma unroll
    for (int i = 0; i < 8; ++i) {
        int t = i * 32 + lane;           // 0..255 -> row, float4 col
        int row = t >> 4, c4 = (t & 15) * 4;
        int qp = qps[row];
        float4 val = make_float4(0.f, 0.f, 0.f, 0.f);
        if (qp >= 0)
            val = *(const float4*)&Q[(((size_t)b * LQ_DIM + qp) * H_DIM + h) * D_DIM + c4];
        *(float4*)&Qs[row][c4] = val;
    }
    __syncthreads();

    // Preload Q in WMMA A-layout: 16 chunks of 4 along D
    v2f qa[16];
#pragma unroll
    for (int c = 0; c < 16; ++c) {
        qa[c].x = Qs[lr][4 * c + 2 * g];
        qa[c].y = Qs[lr][4 * c + 2 * g + 1];
    }
    int qv[8];
#pragma unroll
    for (int r = 0; r < 8; ++r) qv[r] = qps[r + 8 * g];

    float m[8], l[8];
    v8f o[4] = {};
#pragma unroll
    for (int r = 0; r < 8; ++r) { m[r] = -1e30f; l[r] = 0.f; }

    const float scale = 0.125f;  // 1/sqrt(64)

    for (int kt = 0; kt < LK_DIM / 16; ++kt) {
        // Async-copy K and V tiles (16 keys x 64 f32) straight into LDS.
        // 8 x B128 per tile, one lane moves 16B per instruction.
#pragma unroll
        for (int i = 0; i < 8; ++i) {
            int t = i * 32 + lane;
            int row = t >> 4, c4 = (t & 15) * 4;
            size_t gb = (((size_t)b * LK_DIM + (kt * 16 + row)) * H_DIM + h) * D_DIM + c4;
            unsigned kdst = lds_off(&Ks[row][c4]);
            unsigned vdst = lds_off(&Vs[row][c4]);
            const float* kp = &K[gb];
            const float* vp = &V[gb];
            asm volatile("global_load_async_to_lds_b128 %0, %1, off"
                         :: "v"(kdst), "v"(kp) : "memory");
            asm volatile("global_load_async_to_lds_b128 %0, %1, off"
                         :: "v"(vdst), "v"(vp) : "memory");
        }
        asm volatile("s_wait_asynccnt 0x0" ::: "memory");
        __syncthreads();

        // S = Q * K^T  (16 WMMA f32 16x16x4 steps over D=64)
        v8f acc = {};
#pragma unroll
        for (int c = 0; c < 16; ++c) {
            v2f bv;
            bv.x = Ks[lr][4 * c + 2 * g];
            bv.y = Ks[lr][4 * c + 2 * g + 1];
            acc = __builtin_amdgcn_wmma_f32_16x16x4_f32(
                false, qa[c], false, bv, (short)0, acc, false, false);
        }

        // Online softmax update (rows live in lane half-groups of 16);
        // row reductions are immediate-offset ds_swizzle XOR butterflies.
        int key = kt * 16 + lr;
        float pr[8], alpha[8];
#pragma unroll
        for (int r = 0; r < 8; ++r) {
            float s = acc[r] * scale;
            if (key > qv[r]) s = -1e30f;          // causal mask
            float rmax = s;
            rmax = fmaxf(rmax, SWZ_XOR(rmax, 8));
            rmax = fmaxf(rmax, SWZ_XOR(rmax, 4));
            rmax = fmaxf(rmax, SWZ_XOR(rmax, 2));
            rmax = fmaxf(rmax, SWZ_XOR(rmax, 1));
            float mnew = fmaxf(m[r], rmax);
            float p = __expf(s - mnew);
            float rsum = p;
            rsum += SWZ_XOR(rsum, 8);
            rsum += SWZ_XOR(rsum, 4);
            rsum += SWZ_XOR(rsum, 2);
            rsum += SWZ_XOR(rsum, 1);
            alpha[r] = __expf(m[r] - mnew);
            l[r] = l[r] * alpha[r] + rsum;
            m[r] = mnew;
            pr[r] = p;
        }
        __syncthreads();
#pragma unroll
        for (int r = 0; r < 8; ++r) Ps[r + 8 * g][lr] = pr[r];
        // rescale running output
#pragma unroll
        for (int c = 0; c < 4; ++c)
#pragma unroll
            for (int r = 0; r < 8; ++r) o[c][r] *= alpha[r];
        __syncthreads();

        // O += P * V  (4 N-chunks x 4 K-chunks of WMMA f32 16x16x4)
#pragma unroll
        for (int c = 0; c < 4; ++c) {
#pragma unroll
            for (int kc = 0; kc < 4; ++kc) {
                v2f pa, bv;
                pa.x = Ps[lr][4 * kc + 2 * g];
                pa.y = Ps[lr][4 * kc + 2 * g + 1];
                bv.x = Vs[4 * kc + 2 * g][c * 16 + lr];
                bv.y = Vs[4 * kc + 2 * g + 1][c * 16 + lr];
                o[c] = __builtin_amdgcn_wmma_f32_16x16x4_f32(
                    false, pa, false, bv, (short)0, o[c], false, false);
            }
        }
        __syncthreads();
    }

    // Finalize and write valid rows: attn_out[bh, u, d]
#pragma unroll
    for (int r = 0; r < 8; ++r) {
        int u = tile * 16 + r + 8 * g;
        if (u < U_TOP && qv[r] >= 0) {
            float inv = 1.0f / l[r];
#pragma unroll
            for (int c = 0; c < 4; ++c)
                attn_out[((size_t)bh * U_TOP + u) * D_DIM + c * 16 + lr] = o[c][r] * inv;
        }
    }
}

// ---------------------------------------------------------------------------
// Kernel 4a: per-(b,h,chunk) column sums of V (phase 1 of parallel scan).
// ---------------------------------------------------------------------------
__global__ __launch_bounds__(64)
void chunk_sum_kernel(const float* __restrict__ V, float* __restrict__ csum) {
    int blk = blockIdx.x;                 // bh*NCHUNK + chunk
    int chunk = blk % NCHUNK;
    int bh = blk / NCHUNK;
    int h = bh % H_DIM;
    int b = bh / H_DIM;
    int d = threadIdx.x;

    const size_t stride = (size_t)H_DIM * D_DIM;
    size_t p = (size_t)b * LK_DIM * stride + (size_t)(chunk * CHUNK) * stride +
               (size_t)h * D_DIM + d;
    float s = 0.f;
    for (int l = 0; l < CHUNK; ++l) { s += V[p]; p += stride; }
    csum[(size_t)blk * D_DIM + d] = s;
}

// ---------------------------------------------------------------------------
// Kernel 4b: cumsum with chunk base offsets (phase 2) + scatter of the
// attention rows whose positions fall inside this chunk.
// ---------------------------------------------------------------------------
__global__ __launch_bounds__(64)
void cumsum_scatter_kernel(const float* __restrict__ V,
                           const float* __restrict__ csum,
                           const int* __restrict__ Mtop,
                           const float* __restrict__ attn_out,
                           float* __restrict__ out) {
    int blk = blockIdx.x;                 // bh*NCHUNK + chunk
    int chunk = blk % NCHUNK;
    int bh = blk / NCHUNK;
    int h = bh % H_DIM;
    int b = bh / H_DIM;
    int d = threadIdx.x;

    float acc = 0.f;
    for (int c = 0; c < chunk; ++c)
        acc += csum[((size_t)bh * NCHUNK + c) * D_DIM + d];

    const size_t stride = (size_t)H_DIM * D_DIM;
    size_t p = (size_t)b * LK_DIM * stride + (size_t)(chunk * CHUNK) * stride +
               (size_t)h * D_DIM + d;
    for (int l = 0; l < CHUNK; ++l) {
        acc += V[p];
        out[p] = acc;
        p += stride;
    }

    // Scatter rows owned by this chunk. Each d-lane owns column d.
    int lo = chunk * CHUNK, hi = lo + CHUNK;
    for (int u = 0; u < U_TOP; ++u) {
        int pos = Mtop[bh * U_TOP + u];
        if (pos >= lo && pos < hi) {
            size_t op = (size_t)b * LK_DIM * stride + (size_t)pos * stride +
                        (size_t)h * D_DIM + d;
            out[op] = attn_out[((size_t)bh * U_TOP + u) * D_DIM + d];
        }
    }
}

// ---------------------------------------------------------------------------
extern "C" void kernel_launch(void* const* d_in, const int* in_sizes, int n_in,
                              void* d_out, int out_size, void* d_ws, size_t ws_size,
                              hipStream_t stream) {
    (void)n_in; (void)out_size; (void)ws_size;
    const float* Q   = (const float*)d_in[0];
    const float* K   = (const float*)d_in[1];
    const float* V   = (const float*)d_in[2];
    const int*   idx = (const int*)d_in[3];
    float* out = (float*)d_out;

    int u_part = in_sizes[3] / LQ_DIM;   // = 40

    char* ws = (char*)d_ws;
    float* M    = (float*)ws;                                     // B*H*LQ f32
    size_t offM = (size_t)B_DIM * H_DIM * LQ_DIM * sizeof(float); // 512 KB
    int*   Mtop = (int*)(ws + offM);                              // B*H*40 i32
    size_t offT = offM + (size_t)B_DIM * H_DIM * U_TOP * sizeof(int);
    float* attn = (float*)(ws + offT);                            // B*H*40*64 f32
    size_t offA = offT + (size_t)B_DIM * H_DIM * U_TOP * D_DIM * sizeof(float);
    float* csum = (float*)(ws + offA);                            // B*H*16*64 f32

    int nscore = B_DIM * H_DIM * LQ_DIM;
    prob_sample_score_kernel<<<(nscore + 255) / 256, 256, 0, stream>>>(
        Q, K, idx, M, u_part);
    topk_kernel<<<B_DIM * H_DIM, 256, 0, stream>>>(M, Mtop);
    sparse_attn_kernel<<<B_DIM * H_DIM * NTILE, 32, 0, stream>>>(
        Q, K, V, Mtop, attn);
    chunk_sum_kernel<<<B_DIM * H_DIM * NCHUNK, 64, 0, stream>>>(V, csum);
    cumsum_scatter_kernel<<<B_DIM * H_DIM * NCHUNK, 64, 0, stream>>>(
        V, csum, Mtop, attn, out);
}